// Block_6803228196877
// MI455X (gfx1250) — compile-verified
//
#include <hip/hip_runtime.h>

typedef __attribute__((ext_vector_type(2))) float v2f;
typedef __attribute__((ext_vector_type(8))) float v8f;

#define D_FEAT 128

// ---------------------------------------------------------------------------
// Degree / normalization kernels
// ---------------------------------------------------------------------------
__global__ __launch_bounds__(256) void deg_init_kernel(float* deg, int n) {
    int i = blockIdx.x * 256 + threadIdx.x;
    if (i < n) deg[i] = 1.0f;  // self-loop contributes 1
}

__global__ __launch_bounds__(256) void deg_edges_kernel(const int* __restrict__ dst,
                                                        float* deg, int e) {
    int i = blockIdx.x * 256 + threadIdx.x;
    if (i < e) atomicAdd(&deg[dst[i]], 1.0f);
}

__global__ __launch_bounds__(256) void deg_rsqrt_kernel(float* deg, int n) {
    int i = blockIdx.x * 256 + threadIdx.x;
    if (i < n) {
        float d = deg[i];
        deg[i] = (d > 0.0f) ? rsqrtf(d) : 0.0f;  // becomes dinv
    }
}

__global__ __launch_bounds__(256) void zero_kernel(float4* p, long n4) {
    long i = (long)blockIdx.x * 256 + threadIdx.x;
    if (i < n4) p[i] = make_float4(0.f, 0.f, 0.f, 0.f);
}

// ---------------------------------------------------------------------------
// GEMM: H[N x 128] = A[N x 128] @ W[128 x 128], fp32 WMMA 16x16x4.
// Block = 256 threads = 8 waves; block computes 16 rows x 128 cols; each wave
// owns one 16x16 tile. A tile staged in LDS (row stride 132 -> conflict-free).
// ---------------------------------------------------------------------------
#define LDA 132
__global__ __launch_bounds__(256)
void gemm128_wmma_kernel(const float* __restrict__ A, const float* __restrict__ W,
                         float* __restrict__ H, int nrows) {
    __shared__ float As[16 * LDA];
    const int tid = threadIdx.x;
    const long row0 = (long)blockIdx.x * 16;

    // cooperative load of 16x128 A tile: 512 float4, 2 per thread
    for (int idx = tid; idx < 512; idx += 256) {
        int r = idx >> 5;          // row 0..15
        int c4 = idx & 31;         // float4 within row
        long gr = row0 + r;
        if (gr >= nrows) gr = nrows - 1;  // clamp (N is a multiple of 16 anyway)
        *(float4*)(As + r * LDA + c4 * 4) =
            *(const float4*)(A + gr * D_FEAT + c4 * 4);
    }
    __syncthreads();

    const int wave  = tid >> 5;        // 0..7 -> column tile
    const int lane  = tid & 31;
    const int col0  = wave * 16;
    const int mrow  = lane & 15;       // A: M index
    const int khalf = (lane >> 4) * 2; // 0 for lanes 0-15, 2 for lanes 16-31
    const int ncol  = lane & 15;       // B/C/D: N index

    v8f acc = {};
    for (int k = 0; k < D_FEAT; k += 4) {
        v2f a, b;
        a.x = As[mrow * LDA + k + khalf + 0];
        a.y = As[mrow * LDA + k + khalf + 1];
        b.x = W[(k + khalf + 0) * D_FEAT + col0 + ncol];
        b.y = W[(k + khalf + 1) * D_FEAT + col0 + ncol];
        acc = __builtin_amdgcn_wmma_f32_16x16x4_f32(
            /*neg_a=*/false, a, /*neg_b=*/false, b,
            /*c_mod=*/(short)0, acc, /*reuse_a=*/false, /*reuse_b=*/false);
    }

    const int mbase = (lane >> 4) * 8;  // C/D layout: lanes 16-31 hold M+8
    for (int r = 0; r < 8; ++r) {
        long m = row0 + mbase + r;
        if (m < nrows) H[m * D_FEAT + col0 + ncol] = acc[r];
    }
}

// ---------------------------------------------------------------------------
// Edge scatter: one wave per edge, each lane handles 4 features.
// agg[dst] += dinv[src]*dinv[dst] * h[src]
// ---------------------------------------------------------------------------
__global__ __launch_bounds__(256)
void scatter_kernel(const int* __restrict__ src, const int* __restrict__ dst,
                    const float* __restrict__ dinv, const float* __restrict__ h,
                    float* __restrict__ agg, int nedges) {
    int e = blockIdx.x * 8 + (threadIdx.x >> 5);
    if (e >= nedges) return;
    int lane = threadIdx.x & 31;
    int s = src[e], d = dst[e];
    float norm = dinv[s] * dinv[d];
    float4 hv = *(const float4*)(h + (long)s * D_FEAT + lane * 4);
    float* base = agg + (long)d * D_FEAT + lane * 4;
    atomicAdd(base + 0, norm * hv.x);
    atomicAdd(base + 1, norm * hv.y);
    atomicAdd(base + 2, norm * hv.z);
    atomicAdd(base + 3, norm * hv.w);
}

// ---------------------------------------------------------------------------
// Self-loop + bias + ReLU:  x_out = relu(agg + dinv^2 * h + b)
// ---------------------------------------------------------------------------
__global__ __launch_bounds__(256)
void finish_kernel(const float* __restrict__ agg, const float* __restrict__ h,
                   const float* __restrict__ dinv, const float* __restrict__ b,
                   float* __restrict__ xout, int n) {
    long i = (long)blockIdx.x * 256 + threadIdx.x;
    long total = (long)n * D_FEAT;
    if (i >= total) return;
    int node = (int)(i >> 7);
    int f = (int)(i & (D_FEAT - 1));
    float di = dinv[node];
    float v = agg[i] + di * di * h[i] + b[f];
    xout[i] = v > 0.0f ? v : 0.0f;
}

// ---------------------------------------------------------------------------
// Final JK linear: out[N x 128] = concat(x1,x2)[N x 256] @ linW[256 x 128] + b
// Same WMMA tiling, K = 256; A tile read from x1 (k<128) and x2 (k>=128).
// ---------------------------------------------------------------------------
#define LDAJ 260
__global__ __launch_bounds__(256)
void gemm_jk_wmma_kernel(const float* __restrict__ x1, const float* __restrict__ x2,
                         const float* __restrict__ Wl, const float* __restrict__ bl,
                         float* __restrict__ out, int nrows) {
    __shared__ float As[16 * LDAJ];
    const int tid = threadIdx.x;
    const long row0 = (long)blockIdx.x * 16;

    // 16 rows x 256 cols = 1024 float4, 4 per thread
    for (int idx = tid; idx < 1024; idx += 256) {
        int r = idx >> 6;          // row 0..15
        int c4 = idx & 63;         // float4 within 256-wide row
        long gr = row0 + r;
        if (gr >= nrows) gr = nrows - 1;
        const float* srcp = (c4 < 32)
            ? (x1 + gr * D_FEAT + c4 * 4)
            : (x2 + gr * D_FEAT + (c4 - 32) * 4);
        *(float4*)(As + r * LDAJ + c4 * 4) = *(const float4*)srcp;
    }
    __syncthreads();

    const int wave  = tid >> 5;
    const int lane  = tid & 31;
    const int col0  = wave * 16;
    const int mrow  = lane & 15;
    const int khalf = (lane >> 4) * 2;
    const int ncol  = lane & 15;

    v8f acc = {};
    for (int k = 0; k < 2 * D_FEAT; k += 4) {
        v2f a, b;
        a.x = As[mrow * LDAJ + k + khalf + 0];
        a.y = As[mrow * LDAJ + k + khalf + 1];
        b.x = Wl[(k + khalf + 0) * D_FEAT + col0 + ncol];
        b.y = Wl[(k + khalf + 1) * D_FEAT + col0 + ncol];
        acc = __builtin_amdgcn_wmma_f32_16x16x4_f32(
            false, a, false, b, (short)0, acc, false, false);
    }

    const float bias = bl[col0 + ncol];
    const int mbase = (lane >> 4) * 8;
    for (int r = 0; r < 8; ++r) {
        long m = row0 + mbase + r;
        if (m < nrows) out[m * D_FEAT + col0 + ncol] = acc[r] + bias;
    }
}

// ---------------------------------------------------------------------------
// Host orchestration
// ---------------------------------------------------------------------------
extern "C" void kernel_launch(void* const* d_in, const int* in_sizes, int n_in,
                              void* d_out, int out_size, void* d_ws, size_t ws_size,
                              hipStream_t stream) {
    (void)n_in; (void)out_size; (void)ws_size;

    const float* x    = (const float*)d_in[0];
    const int*   ei   = (const int*)d_in[1];
    const float* W1   = (const float*)d_in[2];
    const float* b1   = (const float*)d_in[3];
    const float* W2   = (const float*)d_in[4];
    const float* b2   = (const float*)d_in[5];
    const float* linW = (const float*)d_in[6];
    const float* linb = (const float*)d_in[7];
    float* out = (float*)d_out;

    const int N = in_sizes[0] / D_FEAT;   // 50000
    const int E = in_sizes[1] / 2;        // 800000
    const int* srcI = ei;
    const int* dstI = ei + E;

    // workspace carving (floats)
    float* ws   = (float*)d_ws;
    float* dinv = ws;                        // N
    float* h    = dinv + N;                  // N*128
    float* agg  = h   + (long)N * D_FEAT;    // N*128
    float* x1   = agg + (long)N * D_FEAT;    // N*128
    float* x2   = x1  + (long)N * D_FEAT;    // N*128

    const int nBlkN   = (N + 255) / 256;
    const int nBlkE   = (E + 255) / 256;
    const int nBlkRow = (N + 15) / 16;           // GEMM row blocks
    const int nBlkSc  = (E + 7) / 8;             // scatter: 8 edges/block
    const long feats  = (long)N * D_FEAT;
    const int nBlkF   = (int)((feats + 255) / 256);
    const long n4     = feats / 4;
    const int nBlkZ   = (int)((n4 + 255) / 256);

    // --- degrees (shared by both layers) ---
    deg_init_kernel<<<nBlkN, 256, 0, stream>>>(dinv, N);
    deg_edges_kernel<<<nBlkE, 256, 0, stream>>>(dstI, dinv, E);
    deg_rsqrt_kernel<<<nBlkN, 256, 0, stream>>>(dinv, N);

    // --- layer 1 ---
    gemm128_wmma_kernel<<<nBlkRow, 256, 0, stream>>>(x, W1, h, N);
    zero_kernel<<<nBlkZ, 256, 0, stream>>>((float4*)agg, n4);
    scatter_kernel<<<nBlkSc, 256, 0, stream>>>(srcI, dstI, dinv, h, agg, E);
    finish_kernel<<<nBlkF, 256, 0, stream>>>(agg, h, dinv, b1, x1, N);

    // --- layer 2 ---
    gemm128_wmma_kernel<<<nBlkRow, 256, 0, stream>>>(x1, W2, h, N);
    zero_kernel<<<nBlkZ, 256, 0, stream>>>((float4*)agg, n4);
    scatter_kernel<<<nBlkSc, 256, 0, stream>>>(srcI, dstI, dinv, h, agg, E);
    finish_kernel<<<nBlkF, 256, 0, stream>>>(agg, h, dinv, b2, x2, N);

    // --- JumpingKnowledge concat + linear ---
    gemm_jk_wmma_kernel<<<nBlkRow, 256, 0, stream>>>(x1, x2, linW, linb, out, N);
}